// LocalMatcher_12000138625306
// MI455X (gfx1250) — compile-verified
//
#include <hip/hip_runtime.h>

typedef __attribute__((ext_vector_type(2))) float v2f;
typedef __attribute__((ext_vector_type(8))) float v8f;

#define HIDDEN 64
#define NPOS   32
#define NPATCH 32
#define CIN    4
#define HH     10
#define WW     8
#define PIX    80      // HH*WW = 5 * 16
#define KDIM   36      // CIN*3*3 = 9 * 4

// ---------------------------------------------------------------------------
// Stage 1: both encoders. One wave (32 threads) per image; 64 blocks total.
// blocks 0..31  -> video encoder (split_video image b), writes emb row b
// blocks 32..63 -> patch encoder (p_tensor image b-32), writes emb row b
// Conv as implicit GEMM with V_WMMA_F32_16X16X4_F32: M=80, K=36, N=64.
// ---------------------------------------------------------------------------
__global__ __launch_bounds__(32)
void lm_encode_kernel(const float* __restrict__ v_latent,  // (4,4,10,8,8)
                      const float* __restrict__ p_tensor,  // (32,4,10,8)
                      const float* __restrict__ v_w,       // (64,4,3,3)
                      const float* __restrict__ v_b,       // (64)
                      const float* __restrict__ p_w,       // (64,4,3,3)
                      const float* __restrict__ p_b,       // (64)
                      float* __restrict__ emb)             // (64,64): v rows 0..31, p rows 32..63
{
    __shared__ float sA[PIX][KDIM];     // im2col patches, 80x36 f32
    __shared__ float sB[KDIM][HIDDEN];  // K-major weights, 36x64 f32

    const int lane = threadIdx.x;       // 0..31 (wave32)
    const int half = lane >> 4;         // 0 or 1
    const int lm   = lane & 15;

    const bool isP = (blockIdx.x >= NPOS);
    const int  img = blockIdx.x & (NPOS - 1);

    const float* w    = isP ? p_w : v_w;
    const float* bias = isP ? p_b : v_b;

    // ---- stage weights K-major into LDS: sB[c*9+ky*3+kx][n] = w[n][c][ky][kx]
    for (int idx = lane; idx < KDIM * HIDDEN; idx += 32) {
        int kidx = idx / HIDDEN;
        int n    = idx - kidx * HIDDEN;
        int c    = kidx / 9;
        int r    = kidx - c * 9;        // ky*3+kx
        sB[kidx][n] = w[((n * CIN + c) * 3 + (r / 3)) * 3 + (r % 3)];
    }

    // ---- build im2col patches (zero-padded 3x3 neighborhoods)
    // v image n: v_img[c][h][x] = v_latent[c][n/8][h][n%8][x]   (split_video)
    // p image n: p_tensor[n][c][h][x]
    const int vb = img >> 3;            // n/8
    const int vd = img & 7;             // n%8
    for (int idx = lane; idx < PIX * KDIM; idx += 32) {
        int p    = idx / KDIM;
        int kidx = idx - p * KDIM;
        int y    = p / WW;
        int x    = p - y * WW;
        int c    = kidx / 9;
        int r    = kidx - c * 9;
        int iy   = y + r / 3 - 1;
        int ix   = x + r % 3 - 1;
        float val = 0.0f;
        if (iy >= 0 && iy < HH && ix >= 0 && ix < WW) {
            val = isP ? p_tensor[((img * CIN + c) * HH + iy) * WW + ix]
                      : v_latent[(((c * 4 + vb) * HH + iy) * 8 + vd) * 8 + ix];
        }
        sA[p][kidx] = val;
    }
    __syncthreads();

    // ---- GEMM + fused bias/ReLU/mean-pool
    // Accumulate per-channel ReLU sums; lane's channel for tile nt is nt*16+lm,
    // and it owns pixel rows {mt*16 + i + 8*half : i=0..7}.
    float csum[4] = {0.f, 0.f, 0.f, 0.f};

    for (int nt = 0; nt < 4; ++nt) {
        const float bn = bias[nt * 16 + lm];
        for (int mt = 0; mt < 5; ++mt) {
            v8f acc = {};
            #pragma unroll
            for (int kt = 0; kt < 9; ++kt) {
                const int k0 = kt * 4 + half * 2;
                v2f a, b;
                a.x = sA[mt * 16 + lm][k0];
                a.y = sA[mt * 16 + lm][k0 + 1];
                b.x = sB[k0][nt * 16 + lm];
                b.y = sB[k0 + 1][nt * 16 + lm];
                acc = __builtin_amdgcn_wmma_f32_16x16x4_f32(
                        false, a, false, b, (short)0, acc, false, false);
            }
            #pragma unroll
            for (int i = 0; i < 8; ++i) {
                float v = acc[i] + bn;
                csum[nt] += (v > 0.f) ? v : 0.f;
            }
        }
    }

    // combine the two lane-halves (same channel) and write mean
    const int row = isP ? (NPOS + img) : img;
    #pragma unroll
    for (int nt = 0; nt < 4; ++nt) {
        float tot = csum[nt] + __shfl_xor(csum[nt], 16, 32);
        if (half == 0)
            emb[row * HIDDEN + nt * 16 + lm] = tot * (1.0f / (float)PIX);
    }
}

// ---------------------------------------------------------------------------
// Stage 2: scores (32x32x64 fp32 WMMA GEMM) + stable descending rank + greedy.
// Single block, single wave.
// ---------------------------------------------------------------------------
__global__ __launch_bounds__(32)
void lm_match_kernel(const float* __restrict__ emb,  // (64,64)
                     int* __restrict__ out)          // (32)
{
    __shared__ float sS[NPATCH][NPOS + 1];  // scores, padded stride
    __shared__ int   sPref[NPATCH][NPOS];
    __shared__ int   sAssign[NPATCH];
    __shared__ int   sTaken[NPOS];

    const int lane = threadIdx.x;
    const int half = lane >> 4;
    const int lm   = lane & 15;

    const float* p_emb = emb + NPOS * HIDDEN;  // rows 32..63
    const float* v_emb = emb;                  // rows 0..31

    // scores[m][n] = sum_k p_emb[m][k] * v_emb[n][k]
    // A frag: A[m][k] = p_emb[...]; B frag: B[k][n] = v_emb[n][k]
    for (int mt = 0; mt < 2; ++mt) {
        for (int nt = 0; nt < 2; ++nt) {
            v8f acc = {};
            #pragma unroll
            for (int kt = 0; kt < 16; ++kt) {
                const int k0 = kt * 4 + half * 2;
                v2f a, b;
                a.x = p_emb[(mt * 16 + lm) * HIDDEN + k0];
                a.y = p_emb[(mt * 16 + lm) * HIDDEN + k0 + 1];
                b.x = v_emb[(nt * 16 + lm) * HIDDEN + k0];
                b.y = v_emb[(nt * 16 + lm) * HIDDEN + k0 + 1];
                acc = __builtin_amdgcn_wmma_f32_16x16x4_f32(
                        false, a, false, b, (short)0, acc, false, false);
            }
            #pragma unroll
            for (int i = 0; i < 8; ++i)
                sS[mt * 16 + i + 8 * half][nt * 16 + lm] = acc[i];
        }
    }
    __syncthreads();

    // Stable descending argsort per row (rank-counting, tie -> lower index first),
    // matching jnp.argsort(-scores) stable semantics. Thread t owns row t.
    {
        const int t = lane;
        for (int i = 0; i < NPOS; ++i) {
            const float si = sS[t][i];
            int rank = 0;
            for (int j = 0; j < NPOS; ++j) {
                const float sj = sS[t][j];
                rank += (sj > si) || (sj == si && j < i);
            }
            sPref[t][rank] = i;
        }
        sAssign[t] = -1;
        if (t < NPOS) sTaken[t] = 0;
    }
    __syncthreads();

    // Sequential greedy assignment (exactly mirrors the fori_loop in reference)
    if (lane == 0) {
        for (int i = 0; i < NPOS * NPATCH; ++i) {
            const int pid = i & (NPATCH - 1);
            const int pos = sPref[pid][i >> 5];
            if (sAssign[pid] < 0 && sTaken[pos] == 0) {
                sAssign[pid] = pos;
                sTaken[pos]  = 1;
            }
        }
        for (int p = 0; p < NPATCH; ++p) out[p] = sAssign[p];
    }
}

extern "C" void kernel_launch(void* const* d_in, const int* in_sizes, int n_in,
                              void* d_out, int out_size, void* d_ws, size_t ws_size,
                              hipStream_t stream) {
    const float* v_latent = (const float*)d_in[0];  // (4,4,10,8,8)
    const float* p_tensor = (const float*)d_in[1];  // (32,4,10,8)
    // d_in[2] = p_old_idx : unused by the reference computation
    const float* v_w = (const float*)d_in[3];       // (64,4,3,3)
    const float* v_b = (const float*)d_in[4];       // (64)
    const float* p_w = (const float*)d_in[5];       // (64,4,3,3)
    const float* p_b = (const float*)d_in[6];       // (64)

    float* emb = (float*)d_ws;                      // 64 x 64 f32 = 16 KB

    lm_encode_kernel<<<64, 32, 0, stream>>>(v_latent, p_tensor, v_w, v_b, p_w, p_b, emb);
    lm_match_kernel<<<1, 32, 0, stream>>>(emb, (int*)d_out);
}